// TextFeatureExpander_75986561401268
// MI455X (gfx1250) — compile-verified
//
#include <hip/hip_runtime.h>
#include <stdint.h>

// Length regulator: expand [B,L,D] phoneme features to [B,T,D] frames.
// B=16, L=512, D=256, T=8192 (fixed by the reference setup).
//
// Pure data movement (no FLOPs): store-bound at ~128 MiB out + ~8 MiB unique in.
// Strategy: TDM (tensor_load_to_lds) stages the duration row into LDS,
// LDS scan builds cumsum, binary-search per frame, then b128-coalesced gather.

typedef uint32_t u32x4 __attribute__((ext_vector_type(4)));
typedef uint32_t u32x8 __attribute__((ext_vector_type(8)));

static constexpr int kB = 16;
static constexpr int kL = 512;
static constexpr int kD = 256;
static constexpr int kT = 8192;
static constexpr int kFPB = 64;       // frames per block
static constexpr int kThreads = 256;  // 8 waves (wave32)
static constexpr int kVecPerRow = kD / 4;  // 64 float4 per phoneme row

__global__ __launch_bounds__(kThreads)
void lenreg_expand_kernel(const float* __restrict__ text,
                          const int* __restrict__ dur,
                          float* __restrict__ out)
{
  __shared__ int s_a[kL];       // raw durations (TDM target), then scan ping
  __shared__ int s_b[kL];       // scan pong
  __shared__ int s_idx[kFPB];   // per-frame phoneme index (-1 = zero frame)

  const int tid = threadIdx.x;
  const int b   = blockIdx.y;
  const int t0  = blockIdx.x * kFPB;

  // ---- Stage durations[b, 0..511] (int32) into LDS via the Tensor Data Mover.
  // TDM ignores EXEC and issues per-wave, so only wave 0 takes this branch.
  if (tid < 32) {
    const uint64_t ga = (uint64_t)(uintptr_t)(dur + (size_t)b * kL);
    const uint32_t lds_base = (uint32_t)(uintptr_t)(void*)&s_a[0];

    u32x4 g0;
    g0.x = 1u;                                   // count=1, is_restore=0, no gather
    g0.y = lds_base;                             // lds_addr (bytes)
    g0.z = (uint32_t)ga;                         // global_addr[31:0]
    g0.w = (uint32_t)(ga >> 32) | (2u << 30);    // global_addr[56:32], type=2

    u32x8 g1;
    g1.s0 = 2u << 16;                            // workgroup_mask=0, data_size=2 (4B)
    g1.s1 = ((uint32_t)kL & 0xFFFFu) << 16;      // tensor_dim0[15:0] = 512
    g1.s2 = 1u << 16;                            // tensor_dim0[31:16]=0 | tensor_dim1 lo16 = 1
    g1.s3 = ((uint32_t)kL) << 16;                // tensor_dim1 hi16=0 | tile_dim0 = 512
    g1.s4 = 1u;                                  // tile_dim1=1, tile_dim2=0
    g1.s5 = (uint32_t)kL;                        // tensor_dim0_stride[31:0] = 512
    g1.s6 = 0u;                                  // stride hi / dim1_stride lo
    g1.s7 = 0u;

    // 2-group form (VADDR2/3 = NULL): tensors up to 2D.
    asm volatile("tensor_load_to_lds %0, %1" :: "s"(g0), "s"(g1) : "memory");
#if defined(__has_builtin) && __has_builtin(__builtin_amdgcn_s_wait_tensorcnt)
    __builtin_amdgcn_s_wait_tensorcnt(0);
#else
    asm volatile("s_wait_tensorcnt 0x0" ::: "memory");
#endif
  }
  __syncthreads();

  // ---- Inclusive scan of 512 durations (Hillis-Steele, ping-pong in LDS).
  int* src = s_a;
  int* dst = s_b;
#pragma unroll 1
  for (int s = 1; s < kL; s <<= 1) {
#pragma unroll
    for (int k = 0; k < kL / kThreads; ++k) {
      const int i = tid + k * kThreads;
      int v = src[i];
      if (i >= s) v += src[i - s];
      dst[i] = v;
    }
    __syncthreads();
    int* tmp = src; src = dst; dst = tmp;
  }
  const int total = src[kL - 1];

  // ---- One binary search per frame: first p with cum[p] > t; clamp to L-1.
  if (tid < kFPB) {
    const int t = t0 + tid;
    int lo = 0, hi = kL;
    while (lo < hi) {
      const int mid = (lo + hi) >> 1;
      if (src[mid] > t) hi = mid; else lo = mid + 1;
    }
    int p = (lo < kL) ? lo : (kL - 1);
    if (t < total) {
      s_idx[tid] = p;
      // Warm L2/L0 for the gathered row (emits global_prefetch_b8).
      __builtin_prefetch(text + ((size_t)(b * kL + p)) * kD, 0, 0);
    } else {
      s_idx[tid] = -1;
    }
  }
  __syncthreads();

  // ---- Bulk copy: 64 frames x 64 float4 = 4096 b128 transfers per block.
  // Lane group of 32 covers one 512B-contiguous half-row -> fully coalesced.
  const float4* tf = (const float4*)text;
  float4* o = (float4*)out;
  const size_t srcRowBase = (size_t)b * kL;
  const size_t outBase = ((size_t)b * kT + t0) * kVecPerRow;

#pragma unroll
  for (int j = 0; j < (kFPB * kVecPerRow) / kThreads; ++j) {  // 16 iters
    const int e = j * kThreads + tid;
    const int f = e >> 6;        // frame within tile (kVecPerRow == 64)
    const int c = e & 63;        // float4 column within row
    const int p = s_idx[f];
    float4 v = {0.f, 0.f, 0.f, 0.f};
    if (p >= 0) v = tf[(srcRowBase + (size_t)p) * kVecPerRow + c];
    o[outBase + (size_t)f * kVecPerRow + c] = v;
  }
}

extern "C" void kernel_launch(void* const* d_in, const int* in_sizes, int n_in,
                              void* d_out, int out_size, void* d_ws, size_t ws_size,
                              hipStream_t stream) {
  (void)in_sizes; (void)n_in; (void)out_size; (void)d_ws; (void)ws_size;
  const float* text = (const float*)d_in[0];  // [16,512,256] f32
  const int*   dur  = (const int*)d_in[1];    // [16,512] int
  float* out = (float*)d_out;                 // [16,8192,256] f32

  dim3 grid(kT / kFPB, kB);   // 128 x 16 blocks
  lenreg_expand_kernel<<<grid, kThreads, 0, stream>>>(text, dur, out);
}